// DiscriminatorQuantumCircuit_29128468201910
// MI455X (gfx1250) — compile-verified
//
#include <hip/hip_runtime.h>

// DiscriminatorQuantumCircuit — closed form.
//
// Reference circuit: RY(x_w) encoding on 4 qubits, then 2 layers of
// [CNOT ring ; RZ(weights)] and <Z_0> measurement, sigmoid((z+1)/2).
//
// CNOTs are computational-basis permutations and RZs are diagonal phases,
// both diagonal-compatible with the Z0 measurement, so they only permute
// probabilities. Tracking bit0 through the 8 CNOTs: final bit0 = b0^b1^b3.
// Post-RY state is a product state, so
//     z = cos(x0) * cos(x1) * cos(x3)           (weights & wire-2 cancel)
//     out = sigmoid((z + 1) * 0.5)
//
// Memory-bound streaming kernel: 16 B in (global_load_b128), 4 B out per
// sample; ~0.45 us at 23.3 TB/s HBM (faster from the 192 MB L2 on replays).

__global__ __launch_bounds__(256) void qcirc_closed_form_kernel(
    const float4* __restrict__ in,   // [n] of (x0,x1,x2,x3) per sample
    float* __restrict__ out,         // [n]
    int n) {
  int i = blockIdx.x * blockDim.x + threadIdx.x;
  if (i >= n) return;

  // One aligned 16-byte load per sample -> global_load_b128, 512 B / wave32.
  float4 v = in[i];

  // z = <Z0> = cos(x0)*cos(x1)*cos(x3); wire 2 and all RZ weights cancel.
  float z = __builtin_cosf(v.x) * __builtin_cosf(v.y) * __builtin_cosf(v.w);

  // sigmoid((z + 1) * 0.5)
  float t = (z + 1.0f) * 0.5f;
  out[i] = 1.0f / (1.0f + __builtin_expf(-t));
}

extern "C" void kernel_launch(void* const* d_in, const int* in_sizes, int n_in,
                              void* d_out, int out_size, void* d_ws, size_t ws_size,
                              hipStream_t stream) {
  // d_in[0]: inputs  [BATCH,4] float32  (row-major -> float4 per sample)
  // d_in[1]: weights [2,4]     float32  (provably unused by the output)
  const float4* in = (const float4*)d_in[0];
  float* out = (float*)d_out;

  int n = in_sizes[0] / 4;          // BATCH = 524288
  const int block = 256;            // 8 wave32s
  int grid = (n + block - 1) / block;

  qcirc_closed_form_kernel<<<grid, block, 0, stream>>>(in, out, n);
}